// TF_NP_GRAPH_OPD_53137335386171
// MI455X (gfx1250) — compile-verified
//
#include <hip/hip_runtime.h>

typedef __attribute__((ext_vector_type(2))) float v2f;
typedef __attribute__((ext_vector_type(8))) float v8f;

#define N_STARS 2048
#define N_GRAPH 128
#define G_FEAT  6
#define OPD     256
#define BATCH   1024
#define OPD2    (OPD * OPD)   // 65536
#define KPAD    8             // K=6 padded to 8 for two 16x16x4 WMMAs

// ---------------------------------------------------------------------------
// Kernel 1: one wave32 per query position.
//  - strided scan of obs_pos; match = (x equal) || (y equal); keep min index
//  - butterfly min-reduction reproduces argmax-of-first-True (all-false -> 0)
//  - lane-parallel 128x6 projection graph_dic[idx] @ alpha, butterfly sum
//  - write intermediate row padded to 8 floats (f=6,7 = 0) into workspace
// ---------------------------------------------------------------------------
__global__ __launch_bounds__(256) void tfnp_index_project(
    const float* __restrict__ positions,   // (BATCH, 2)
    const float* __restrict__ obs_pos,     // (N_STARS, 2)
    const float* __restrict__ graph_dic,   // (N_STARS, N_GRAPH)
    const float* __restrict__ alpha,       // (N_GRAPH, G_FEAT)
    float* __restrict__ inter)             // (BATCH, KPAD) workspace
{
    const int lane = threadIdx.x & 31;
    const int b    = blockIdx.x * 8 + (threadIdx.x >> 5);

    const float px = positions[2 * b + 0];
    const float py = positions[2 * b + 1];

    int best = N_STARS;  // sentinel: larger than any valid index
    for (int i = lane; i < N_STARS; i += 32) {
        const float ox = obs_pos[2 * i + 0];
        const float oy = obs_pos[2 * i + 1];
        if (ox == px || oy == py) best = min(best, i);
    }
#pragma unroll
    for (int off = 16; off > 0; off >>= 1)
        best = min(best, __shfl_xor(best, off, 32));
    const int idx = (best == N_STARS) ? 0 : best;  // argmax(all False) == 0

    const float* __restrict__ g = graph_dic + (size_t)idx * N_GRAPH;
    float acc[G_FEAT];
#pragma unroll
    for (int f = 0; f < G_FEAT; ++f) acc[f] = 0.0f;
    for (int k = lane; k < N_GRAPH; k += 32) {
        const float gv = g[k];
        const float* __restrict__ arow = alpha + k * G_FEAT;
#pragma unroll
        for (int f = 0; f < G_FEAT; ++f) acc[f] = fmaf(gv, arow[f], acc[f]);
    }
#pragma unroll
    for (int f = 0; f < G_FEAT; ++f) {
#pragma unroll
        for (int off = 16; off > 0; off >>= 1)
            acc[f] += __shfl_xor(acc[f], off, 32);
    }
    if (lane < KPAD)
        inter[b * KPAD + lane] = (lane < G_FEAT) ? acc[lane] : 0.0f;
}

// ---------------------------------------------------------------------------
// Kernel 2: out[b, p] = sum_f inter[b, f] * S[f, p]
//   GEMM  M=1024 (batch) x N=65536 (pixels) x K=8 (padded features)
//   via two V_WMMA_F32_16X16X4_F32 per 16x16 tile. Store-bandwidth bound:
//   256 MB written once -> non-temporal stores.
//
// A layout (16x4 f32, 2 VGPRs): lanes 0-15 K={0,1}, lanes 16-31 K={2,3}.
// B layout (4x16, 2 VGPRs):     VGPR0 rows {0|2}, VGPR1 rows {1|3} per half-wave.
// A's K=6,7 entries are exact zeros (padded inter), so the B values loaded for
// the upper half-wave of the second WMMA (re-read rows 4,5; finite) are
// annihilated — no divergence, no out-of-bounds reads of S.
// ---------------------------------------------------------------------------
__global__ __launch_bounds__(256) void tfnp_opd_wmma(
    const float* __restrict__ inter,  // (BATCH, KPAD)
    const float* __restrict__ S,      // (G_FEAT, OPD2)
    float* __restrict__ out)          // (BATCH, OPD2)
{
    const int lane = threadIdx.x & 31;
    const int wave = threadIdx.x >> 5;
    const int rb   = blockIdx.y;       // 16-row block of the batch dimension
    const int m    = lane & 15;
    const int hi   = lane >> 4;        // 0: lanes 0-15, 1: lanes 16-31

    // A tiles (reused across all column tiles of this wave)
    const float* __restrict__ arow = inter + (rb * 16 + m) * KPAD;
    v2f a_lo, a_hi;
    a_lo.x = arow[2 * hi + 0];   // K = 0 | 2
    a_lo.y = arow[2 * hi + 1];   // K = 1 | 3
    a_hi.x = arow[2 * hi + 4];   // K = 4 | 6 (zero)
    a_hi.y = arow[2 * hi + 5];   // K = 5 | 7 (zero)

    const float* __restrict__ srow0 = S + (size_t)(2 * hi) * OPD2;  // row 0 | 2
    const float* __restrict__ srow1 = srow0 + OPD2;                 // row 1 | 3
    const float* __restrict__ srow4 = S + (size_t)4 * OPD2;         // row 4 (both halves)
    const float* __restrict__ srow5 = srow4 + OPD2;                 // row 5 (both halves)

    const long out_base = (long)(rb * 16 + 8 * hi) * OPD2;
    const int  nt0      = (blockIdx.x * 8 + wave) * 8;  // 8 column tiles/wave

    for (int i = 0; i < 8; ++i) {
        const int col = (nt0 + i) * 16 + m;
        v2f b_lo, b_hi;
        b_lo.x = srow0[col];
        b_lo.y = srow1[col];
        b_hi.x = srow4[col];   // upper half-wave: dead values, A zeros kill them
        b_hi.y = srow5[col];

        v8f c = {};
        c = __builtin_amdgcn_wmma_f32_16x16x4_f32(false, a_lo, false, b_lo,
                                                  (short)0, c, false, false);
        c = __builtin_amdgcn_wmma_f32_16x16x4_f32(false, a_hi, false, b_hi,
                                                  (short)0, c, false, false);

        float* __restrict__ op = out + out_base + col;
#pragma unroll
        for (int r = 0; r < 8; ++r)
            __builtin_nontemporal_store(c[r], op + (long)r * OPD2);
    }
}

extern "C" void kernel_launch(void* const* d_in, const int* in_sizes, int n_in,
                              void* d_out, int out_size, void* d_ws, size_t ws_size,
                              hipStream_t stream) {
    const float* positions = (const float*)d_in[0];  // (1024, 2)
    const float* obs_pos   = (const float*)d_in[1];  // (2048, 2)
    const float* graph_dic = (const float*)d_in[2];  // (2048, 128)
    const float* alpha     = (const float*)d_in[3];  // (128, 6)
    const float* S_graph   = (const float*)d_in[4];  // (6, 256, 256)
    float*       out       = (float*)d_out;          // (1024, 256, 256)
    float*       inter     = (float*)d_ws;           // (1024, 8) = 32 KB scratch

    // Kernel 1: 1024 waves (8 per 256-thread block)
    tfnp_index_project<<<BATCH / 8, 256, 0, stream>>>(
        positions, obs_pos, graph_dic, alpha, inter);

    // Kernel 2: 64 row-blocks x 64 column-chunks, 8 waves/block,
    // 8 tiles/wave -> 4096 column tiles x 64 row blocks = all 262144 tiles.
    tfnp_opd_wmma<<<dim3(64, 64), 256, 0, stream>>>(inter, S_graph, out);
}